// LabelGCN_60447369723925
// MI455X (gfx1250) — compile-verified
//
#include <hip/hip_runtime.h>

typedef float v2f __attribute__((ext_vector_type(2)));
typedef float v8f __attribute__((ext_vector_type(8)));

#define CS_EPS 1e-8f
#define WAVES 4
#define TPB (WAVES * 32)
#define FLT_BIG 3.402823466e+38f

__device__ __forceinline__ void lds_fence() {
  // wave-private LDS region + lockstep wave32: waiting on DScnt==0 makes all
  // prior ds_store results visible to every lane of this wave.
  asm volatile("s_wait_dscnt 0" ::: "memory");
}

__device__ __forceinline__ float relu1(float v) {
  // single v_med3_f32 (median of {v,0,+big}) instead of canonicalize+max
  return __builtin_amdgcn_fmed3f(v, 0.0f, FLT_BIG);
}

__device__ __forceinline__ v8f splat8(float v) {
  v8f r;
#pragma unroll
  for (int i = 0; i < 8; ++i) r[i] = v;
  return r;
}

__device__ __forceinline__ v8f wmma4(v2f a, v2f b, v8f c) {
  // V_WMMA_F32_16X16X4_F32 : D[16x16] = A[16x4] * B[4x16] + C
  return __builtin_amdgcn_wmma_f32_16x16x4_f32(
      false, a, false, b, (short)0, c, false, false);
}

// ---------------------------------------------------------------------------
// Kernel 1: union over batch of thresholded cosine-similarity adjacency.
// ---------------------------------------------------------------------------
__global__ void adj_union_kernel(const float* __restrict__ x,
                                 unsigned* __restrict__ mask, int nSamp) {
  int b = blockIdx.x * blockDim.x + threadIdx.x;
  unsigned m = 0;
  if (b < nSamp) {
    const float4* xs = (const float4*)x + (size_t)b * 4;
    float4 r[4];
#pragma unroll
    for (int i = 0; i < 4; ++i) r[i] = xs[i];
    float inv[4];
#pragma unroll
    for (int i = 0; i < 4; ++i) {
      float nn = r[i].x * r[i].x + r[i].y * r[i].y + r[i].z * r[i].z + r[i].w * r[i].w;
      inv[i] = 1.0f / fmaxf(sqrtf(nn), CS_EPS);
    }
#pragma unroll
    for (int i = 0; i < 4; ++i) {
#pragma unroll
      for (int j = i + 1; j < 4; ++j) {
        float d = r[i].x * r[j].x + r[i].y * r[j].y + r[i].z * r[j].z + r[i].w * r[j].w;
        if (d * inv[i] * inv[j] > 0.5f) m |= (1u << (i * 4 + j)) | (1u << (j * 4 + i));
      }
    }
  }
  __shared__ unsigned bm;
  if (threadIdx.x == 0) bm = 0;
  __syncthreads();
  if (m) atomicOr(&bm, m);
  __syncthreads();
  if (threadIdx.x == 0 && bm) atomicOr(mask, bm);
}

// ---------------------------------------------------------------------------
// Kernel 2: Ahat = D^-1/2 (A + I) D^-1/2   (4x4, single thread)
// ---------------------------------------------------------------------------
__global__ void build_ahat_kernel(const unsigned* __restrict__ mask,
                                  float* __restrict__ ahat) {
  unsigned mk = *mask;
  float A[16];
  for (int i = 0; i < 4; ++i)
    for (int j = 0; j < 4; ++j)
      A[i * 4 + j] = (i == j) ? 1.0f : (float)((mk >> (i * 4 + j)) & 1u);
  float dinv[4];
  for (int i = 0; i < 4; ++i) {
    float s = A[i * 4 + 0] + A[i * 4 + 1] + A[i * 4 + 2] + A[i * 4 + 3];
    dinv[i] = rsqrtf(s);
  }
  for (int i = 0; i < 4; ++i)
    for (int j = 0; j < 4; ++j)
      ahat[i * 4 + j] = dinv[i] * A[i * 4 + j] * dinv[j];
}

// ---------------------------------------------------------------------------
// Kernel 3: fused MLP (4->128->64->64) + GCN (64->32->4), all matmuls
// (including the block-diag Ahat mixes) via v_wmma_f32_16x16x4_f32.
//
// Weight LDS layout: packed by K-pair  w[p][n][2] with p = k>>1, row stride
// chosen == 32 (mod 64) dwords so the two lane-halves (which read rows p and
// p+1... i.e. p offset by `half`) hit disjoint bank sets; each B tile is one
// aligned ds_load_b64.
// ---------------------------------------------------------------------------
__global__ __launch_bounds__(TPB) void labelgcn_kernel(
    const float* __restrict__ x,
    const float* __restrict__ W1, const float* __restrict__ b1,
    const float* __restrict__ W2, const float* __restrict__ b2,
    const float* __restrict__ W3, const float* __restrict__ b3,
    const float* __restrict__ Wg1, const float* __restrict__ bg1,
    const float* __restrict__ Wg2, const float* __restrict__ bg2,
    const float* __restrict__ ahatG,
    float* __restrict__ embOut, float* __restrict__ gcnOut, int nTiles) {
  __shared__ __align__(16) float sW1p[2 * 288];    // [p<2][n<128][2], stride 288
  __shared__ __align__(16) float sW2p[64 * 160];   // [p<64][n<64][2],  stride 160
  __shared__ __align__(16) float sW3p[32 * 160];   // [p<32][n<64][2],  stride 160
  __shared__ __align__(16) float sWg1p[32 * 96];   // [p<32][n<32][2],  stride 96
  __shared__ __align__(16) float sWg2p[16 * 32];   // [p<16][n<16][2],  stride 32 (zero-padded N)
  __shared__ float sB1[128], sB2[64], sB3[64], sBg1[32], sBg2p[16], sAhat[16];
  // per-wave staging: hA: h1[16x128] -> emb[16x64] -> g[16x32];  hB: h2 -> t1 -> t2
  __shared__ __align__(16) float hA[WAVES][16 * 132];
  __shared__ __align__(16) float hB[WAVES][16 * 68];

  const int tid = threadIdx.x;
  for (int i = tid; i < 4 * 128; i += TPB) {
    int k = i >> 7, n = i & 127;
    sW1p[(k >> 1) * 288 + 2 * n + (k & 1)] = W1[i];
  }
  for (int i = tid; i < 128 * 64; i += TPB) {
    int k = i >> 6, n = i & 63;
    sW2p[(k >> 1) * 160 + 2 * n + (k & 1)] = W2[i];
  }
  for (int i = tid; i < 64 * 64; i += TPB) {
    int k = i >> 6, n = i & 63;
    sW3p[(k >> 1) * 160 + 2 * n + (k & 1)] = W3[i];
  }
  for (int i = tid; i < 64 * 32; i += TPB) {
    int k = i >> 5, n = i & 31;
    sWg1p[(k >> 1) * 96 + 2 * n + (k & 1)] = Wg1[i];
  }
  for (int i = tid; i < 32 * 16; i += TPB) {  // Wg2 zero-padded [32][16]
    int k = i >> 4, n = i & 15;
    sWg2p[(k >> 1) * 32 + 2 * n + (k & 1)] = (n < 4) ? Wg2[k * 4 + n] : 0.0f;
  }
  for (int i = tid; i < 128; i += TPB) sB1[i] = b1[i];
  for (int i = tid; i < 64; i += TPB)  sB2[i] = b2[i];
  for (int i = tid; i < 64; i += TPB)  sB3[i] = b3[i];
  for (int i = tid; i < 32; i += TPB)  sBg1[i] = bg1[i];
  if (tid < 16) sBg2p[tid] = (tid < 4) ? bg2[tid] : 0.0f;
  if (tid < 16) sAhat[tid] = ahatG[tid];
  __syncthreads();

  const int lane = tid & 31;
  const int wv = tid >> 5;
  const int half = lane >> 4;
  const int lm = lane & 15;
  float* A_ = hA[wv];
  float* B_ = hB[wv];

  // Tile-invariant A-operand tiles of blockdiag(Ahat,Ahat,Ahat,Ahat) [16x16]:
  // a_mix[kt] holds columns 4kt..4kt+3; nonzero only when kt == row/4.
  v2f a_mix[4];
#pragma unroll
  for (int kt = 0; kt < 4; ++kt) {
    bool on = ((lm >> 2) == kt);
    a_mix[kt].x = on ? sAhat[(lm & 3) * 4 + 2 * half + 0] : 0.0f;
    a_mix[kt].y = on ? sAhat[(lm & 3) * 4 + 2 * half + 1] : 0.0f;
  }

  const int slot = blockIdx.x * WAVES + wv;
  const int nSlot = gridDim.x * WAVES;

  for (int tile = slot; tile < nTiles; tile += nSlot) {
    const size_t row0 = (size_t)tile * 16;
    lds_fence();

    // ---- layer 1: h1 = relu(x @ W1 + b1), [16,4]x[4,128] -> hA
    v2f a0 = *(const v2f*)(x + (row0 + lm) * 4 + 2 * half);
#pragma unroll
    for (int nt = 0; nt < 8; ++nt) {
      int n = nt * 16 + lm;
      v2f b = *(const v2f*)&sW1p[half * 288 + 2 * n];
      v8f acc = splat8(sB1[n]);
      acc = wmma4(a0, b, acc);
#pragma unroll
      for (int r = 0; r < 8; ++r)
        A_[(r + 8 * half) * 132 + n] = relu1(acc[r]);
    }
    lds_fence();

    // ---- layer 2: h2 = relu(h1 @ W2 + b2), [16,128]x[128,64] -> hB
    {
      v8f acc[4];
#pragma unroll
      for (int nt = 0; nt < 4; ++nt) acc[nt] = splat8(sB2[nt * 16 + lm]);
#pragma unroll
      for (int kt = 0; kt < 32; ++kt) {
        int k0 = 4 * kt + 2 * half;
        int p = 2 * kt + half;
        v2f a = *(const v2f*)&A_[lm * 132 + k0];
#pragma unroll
        for (int nt = 0; nt < 4; ++nt) {
          v2f b = *(const v2f*)&sW2p[p * 160 + 2 * (nt * 16 + lm)];
          acc[nt] = wmma4(a, b, acc[nt]);
        }
      }
#pragma unroll
      for (int nt = 0; nt < 4; ++nt)
#pragma unroll
        for (int r = 0; r < 8; ++r)
          B_[(r + 8 * half) * 68 + nt * 16 + lm] = relu1(acc[nt][r]);
    }
    lds_fence();

    // ---- layer 3: emb = h2 @ W3 + b3 (no relu), [16,64]x[64,64] -> hA
    {
      v8f acc[4];
#pragma unroll
      for (int nt = 0; nt < 4; ++nt) acc[nt] = splat8(sB3[nt * 16 + lm]);
#pragma unroll
      for (int kt = 0; kt < 16; ++kt) {
        int k0 = 4 * kt + 2 * half;
        int p = 2 * kt + half;
        v2f a = *(const v2f*)&B_[lm * 68 + k0];
#pragma unroll
        for (int nt = 0; nt < 4; ++nt) {
          v2f b = *(const v2f*)&sW3p[p * 160 + 2 * (nt * 16 + lm)];
          acc[nt] = wmma4(a, b, acc[nt]);
        }
      }
#pragma unroll
      for (int nt = 0; nt < 4; ++nt)
#pragma unroll
        for (int r = 0; r < 8; ++r)
          A_[(r + 8 * half) * 132 + nt * 16 + lm] = acc[nt][r];
    }
    lds_fence();

    // ---- emb writeback: coalesced b128 stores from staged tile
#pragma unroll
    for (int q = 0; q < 8; ++q) {
      int c = lane + 32 * q;            // 256 chunks of 4 floats
      int rr = c >> 4, col = (c & 15) * 4;
      float4 v4 = *(const float4*)&A_[rr * 132 + col];
      *(float4*)(embOut + (row0 + rr) * 64 + col) = v4;
    }

    // ---- gcn1: t1 = emb @ Wg1, [16,64]x[64,32] -> hB
    {
      v8f acc[2];
#pragma unroll
      for (int nt = 0; nt < 2; ++nt) acc[nt] = splat8(0.0f);
#pragma unroll
      for (int kt = 0; kt < 16; ++kt) {
        int k0 = 4 * kt + 2 * half;
        int p = 2 * kt + half;
        v2f a = *(const v2f*)&A_[lm * 132 + k0];
#pragma unroll
        for (int nt = 0; nt < 2; ++nt) {
          v2f b = *(const v2f*)&sWg1p[p * 96 + 2 * (nt * 16 + lm)];
          acc[nt] = wmma4(a, b, acc[nt]);
        }
      }
#pragma unroll
      for (int nt = 0; nt < 2; ++nt)
#pragma unroll
        for (int r = 0; r < 8; ++r)
          B_[(r + 8 * half) * 68 + nt * 16 + lm] = acc[nt][r];
    }
    lds_fence();

    // ---- mix 1 (WMMA): g = relu(blockdiag(Ahat) @ t1 + bg1) -> hA cols 0..31
#pragma unroll
    for (int nt = 0; nt < 2; ++nt) {
      int n = nt * 16 + lm;
      v8f acc = splat8(sBg1[n]);
#pragma unroll
      for (int kt = 0; kt < 4; ++kt) {
        int k0 = 4 * kt + 2 * half;
        v2f b;
        b.x = B_[(k0 + 0) * 68 + n];
        b.y = B_[(k0 + 1) * 68 + n];
        acc = wmma4(a_mix[kt], b, acc);
      }
#pragma unroll
      for (int r = 0; r < 8; ++r)
        A_[(r + 8 * half) * 132 + n] = relu1(acc[r]);
    }
    lds_fence();

    // ---- gcn2 (WMMA): t2 = g @ Wg2pad, [16,32]x[32,16] -> hB cols 0..15
    {
      v8f acc = splat8(0.0f);
#pragma unroll
      for (int kt = 0; kt < 8; ++kt) {
        int k0 = 4 * kt + 2 * half;
        int p = 2 * kt + half;
        v2f a = *(const v2f*)&A_[lm * 132 + k0];
        v2f b = *(const v2f*)&sWg2p[p * 32 + 2 * lm];
        acc = wmma4(a, b, acc);
      }
#pragma unroll
      for (int r = 0; r < 8; ++r)
        B_[(r + 8 * half) * 68 + lm] = acc[r];
    }
    lds_fence();

    // ---- mix 2 (WMMA): out = blockdiag(Ahat) @ t2 + bg2; store cols 0..3
    {
      v8f acc = splat8(sBg2p[lm]);
#pragma unroll
      for (int kt = 0; kt < 4; ++kt) {
        int k0 = 4 * kt + 2 * half;
        v2f b;
        b.x = B_[(k0 + 0) * 68 + lm];
        b.y = B_[(k0 + 1) * 68 + lm];
        acc = wmma4(a_mix[kt], b, acc);
      }
      if (lm < 4) {
#pragma unroll
        for (int r = 0; r < 8; ++r)
          gcnOut[(row0 + r + 8 * half) * 4 + lm] = acc[r];
      }
    }
  }
}

// ---------------------------------------------------------------------------
extern "C" void kernel_launch(void* const* d_in, const int* in_sizes, int n_in,
                              void* d_out, int out_size, void* d_ws,
                              size_t ws_size, hipStream_t stream) {
  (void)n_in; (void)out_size; (void)ws_size;
  const float* x   = (const float*)d_in[0];
  const float* W1  = (const float*)d_in[1];
  const float* b1  = (const float*)d_in[2];
  const float* W2  = (const float*)d_in[3];
  const float* b2  = (const float*)d_in[4];
  const float* W3  = (const float*)d_in[5];
  const float* b3  = (const float*)d_in[6];
  const float* Wg1 = (const float*)d_in[7];
  const float* bg1 = (const float*)d_in[8];
  const float* Wg2 = (const float*)d_in[9];
  const float* bg2 = (const float*)d_in[10];

  const int nSamp = in_sizes[0] / 16;  // B  (x is [B,4,4])
  const int nRows = nSamp * 4;         // B*C
  const int nTiles = nRows / 16;

  unsigned* mask = (unsigned*)d_ws;
  float* ahat = (float*)((char*)d_ws + 16);

  hipMemsetAsync(d_ws, 0, 16, stream);
  adj_union_kernel<<<(nSamp + 255) / 256, 256, 0, stream>>>(x, mask, nSamp);
  build_ahat_kernel<<<1, 1, 0, stream>>>(mask, ahat);

  float* embOut = (float*)d_out;
  float* gcnOut = embOut + (size_t)nRows * 64;

  int blocks = (nTiles + WAVES - 1) / WAVES;
  if (blocks > 2048) blocks = 2048;
  if (blocks < 1) blocks = 1;
  labelgcn_kernel<<<blocks, TPB, 0, stream>>>(x, W1, b1, W2, b2, W3, b3, Wg1,
                                              bg1, Wg2, bg2, ahat, embOut,
                                              gcnOut, nTiles);
}